// Cross_Transformer2_68341519614500
// MI455X (gfx1250) — compile-verified
//
#include <hip/hip_runtime.h>

// MI455X / gfx1250, wave32. Fused fp32 transformer block using
// V_WMMA_F32_16X16X4_F32 (full-fp32 matrix path) + flash attention with
// double-buffered GLOBAL_LOAD_ASYNC_TO_LDS_B128 K/V staging (ASYNCcnt).

typedef __attribute__((ext_vector_type(2))) float v2f;
typedef __attribute__((ext_vector_type(8))) float v8f;

#define D_MODEL 256
#define H_DIM   128
#define LN_EPS  1e-5f

// D = A(16x4) * B(4x16) + C(16x16), fp32.
// Layout (ISA 7.12.2): A/B component c holds K = c + 2*(lane>=16);
// C/D component r holds row r + 8*(lane>=16), col = lane&15.
static __device__ __forceinline__ v8f wmma4(v2f a, v2f b, v8f c) {
  return __builtin_amdgcn_wmma_f32_16x16x4_f32(
      /*neg_a=*/false, a, /*neg_b=*/false, b,
      /*c_mod=*/(short)0, c, /*reuse_a=*/false, /*reuse_b=*/false);
}

static __device__ __forceinline__ float redmax16(float v) {
  #pragma unroll
  for (int m = 1; m < 16; m <<= 1) v = fmaxf(v, __shfl_xor(v, m, 32));
  return v;  // stays within each 16-lane half (masks < 16)
}
static __device__ __forceinline__ float redsum16(float v) {
  #pragma unroll
  for (int m = 1; m < 16; m <<= 1) v += __shfl_xor(v, m, 32);
  return v;
}

// Async DMA of 16 bytes/lane into LDS, GVS addressing (SADDR = uniform base,
// VADDR = per-lane byte offset), tracked by ASYNCcnt. VDST holds the LDS byte
// address: flat->LDS mapping uses addr[31:0], so truncating the generic
// pointer to 32 bits yields the LDS offset (ISA 10.2).
static __device__ __forceinline__ void async_ld_b128(const float* lds_dst,
                                                     unsigned goff_bytes,
                                                     const float* gbase) {
  unsigned lds_off = (unsigned)(uintptr_t)lds_dst;
  asm volatile("global_load_async_to_lds_b128 %0, %1, %2"
               :: "v"(lds_off), "v"(goff_bytes), "s"(gbase)
               : "memory");
}

// Stage one 16x256 fp32 block (16 KB) into LDS: 16 b128 ops per thread with
// 64 threads -> 16 async instructions per wave.
static __device__ __forceinline__ void stage_async16x256(const float* gsrc,
                                                         float* lds_base,
                                                         int tid) {
  #pragma unroll
  for (int i = 0; i < 16; ++i) {
    const int idx = tid + i * 64;  // float4 index 0..1023
    async_ld_b128(lds_base + idx * 4, (unsigned)(idx * 16), gsrc);
  }
}

// ---------------------------------------------------------------------------
// Y[M,256] = X[M,256] @ W[256,256] + bias.  One 16-row strip per block,
// 8 waves; each wave computes two 16x16 output tiles. W stays L2-resident.
// ---------------------------------------------------------------------------
__global__ __launch_bounds__(256) void proj_gemm_kernel(
    const float* __restrict__ X, const float* __restrict__ W,
    const float* __restrict__ bias, float* __restrict__ Y)
{
  __shared__ float Xs[16 * D_MODEL];  // 16 KB
  const int row0 = blockIdx.x * 16;
  {
    const float4* src = (const float4*)(X + (size_t)row0 * D_MODEL);
    float4* dst = (float4*)Xs;
    for (int i = threadIdx.x; i < 16 * D_MODEL / 4; i += blockDim.x) dst[i] = src[i];
  }
  __syncthreads();

  const int wave = threadIdx.x >> 5, lane = threadIdx.x & 31;
  const int half = lane >> 4, l16 = lane & 15;

  for (int t = 0; t < 2; ++t) {
    const int n0 = wave * 32 + t * 16;
    v8f acc = {};
    #pragma unroll 8
    for (int kk = 0; kk < D_MODEL; kk += 4) {
      const int ka = kk + 2 * half;
      v2f a = *(const v2f*)&Xs[l16 * D_MODEL + ka];         // ds_load_b64
      v2f b;
      b.x = W[(size_t)ka * D_MODEL + n0 + l16];
      b.y = W[(size_t)(ka + 1) * D_MODEL + n0 + l16];
      acc = wmma4(a, b, acc);
    }
    const float bv = bias[n0 + l16];
    #pragma unroll
    for (int r = 0; r < 8; ++r)
      Y[(size_t)(row0 + r + 8 * half) * D_MODEL + n0 + l16] = acc[r] + bv;
  }
}

// ---------------------------------------------------------------------------
// out = LayerNorm(softmax(Q K^T) V + R).  Flash attention: online softmax
// over 16-key blocks; K/V double-buffered in LDS via async DMA so the next
// block streams in while WMMA consumes the current one. 2 waves per block.
// ---------------------------------------------------------------------------
__global__ __launch_bounds__(64) void attn_ln_kernel(
    const float* __restrict__ Q, const float* __restrict__ K,
    const float* __restrict__ V, const float* __restrict__ R,
    const float* __restrict__ gamma, const float* __restrict__ beta,
    float* __restrict__ out, int S)
{
  __shared__ float Ks[2][16 * D_MODEL];  // 2 x 16 KB
  __shared__ float Vs[2][16 * D_MODEL];  // 2 x 16 KB
  __shared__ float Ps[2 * 256];          // per-wave P-tile staging (C->A layout)

  const int tid = threadIdx.x;
  const int wave = tid >> 5, lane = tid & 31;
  const int half = lane >> 4, l16 = lane & 15;
  const int row0 = blockIdx.x * 32 + wave * 16;

  v8f zero = {};
  v8f o[16];                           // 16x256 fp32 accumulator (128 VGPRs)
  #pragma unroll
  for (int t = 0; t < 16; ++t) o[t] = zero;

  float m_r[8], l_r[8];
  #pragma unroll
  for (int r = 0; r < 8; ++r) { m_r[r] = -3.0e38f; l_r[r] = 0.0f; }

  const float* Qrow = Q + (size_t)(row0 + l16) * D_MODEL;  // A-frag base (L0-cached)
  float* Pw = &Ps[wave * 256];

  const int nb = S >> 4;
  // prologue: start DMA of block 0 into buffer 0
  stage_async16x256(K, &Ks[0][0], tid);
  stage_async16x256(V, &Vs[0][0], tid);

  for (int jb = 0; jb < nb; ++jb) {
    const int cur = jb & 1;
    // kick off next block's DMA into the other buffer, then retire current's
    // 32 in-order async ops (wait threshold = ops just issued).
    if (jb + 1 < nb) {
      const float* Kn = K + (size_t)(jb + 1) * 16 * D_MODEL;
      const float* Vn = V + (size_t)(jb + 1) * 16 * D_MODEL;
      stage_async16x256(Kn, &Ks[cur ^ 1][0], tid);
      stage_async16x256(Vn, &Vs[cur ^ 1][0], tid);
      asm volatile("s_wait_asynccnt 0x20" ::: "memory");
    } else {
      asm volatile("s_wait_asynccnt 0x0" ::: "memory");
    }
    __syncthreads();
    const float* Kc = &Ks[cur][0];
    const float* Vc = &Vs[cur][0];

    // logits tile: S16 = Qtile(16x256) @ Kblk^T(256x16)  -> 64 wmma
    v8f s = zero;
    #pragma unroll 8
    for (int kk = 0; kk < D_MODEL; kk += 4) {
      const int ka = kk + 2 * half;
      v2f a = *(const v2f*)&Qrow[ka];
      v2f b = *(const v2f*)&Kc[l16 * D_MODEL + ka];  // B[k][n] = Kblk[n][k]
      s = wmma4(a, b, s);
    }

    // online softmax update (per-row stats via 16-lane shuffles; the lane
    // halves match the C-layout row split, so alpha aligns with o[t][r])
    float alpha[8];
    #pragma unroll
    for (int r = 0; r < 8; ++r) {
      const float mx = redmax16(s[r]);
      const float nm = fmaxf(m_r[r], mx);
      alpha[r] = __expf(m_r[r] - nm);
      const float p = __expf(s[r] - nm);
      s[r] = p;
      l_r[r] = l_r[r] * alpha[r] + redsum16(p);
      m_r[r] = nm;
    }

    // stage P (C-layout) to LDS, read back as A-fragments
    #pragma unroll
    for (int r = 0; r < 8; ++r)
      Pw[(r + 8 * half) * 16 + l16] = s[r];
    asm volatile("s_wait_dscnt 0" ::: "memory");
    v2f pa[4];
    #pragma unroll
    for (int c = 0; c < 4; ++c)
      pa[c] = *(const v2f*)&Pw[l16 * 16 + c * 4 + 2 * half];

    // O = alpha*O + P(16x16) @ Vblk(16x256)  -> 64 wmma
    #pragma unroll
    for (int t = 0; t < 16; ++t) {
      v8f acc = o[t];
      #pragma unroll
      for (int r = 0; r < 8; ++r) acc[r] *= alpha[r];
      #pragma unroll
      for (int c = 0; c < 4; ++c) {
        const int kb = c * 4 + 2 * half;
        v2f b;
        b.x = Vc[(size_t)kb * D_MODEL + t * 16 + l16];
        b.y = Vc[(size_t)(kb + 1) * D_MODEL + t * 16 + l16];
        acc = wmma4(pa[c], b, acc);
      }
      o[t] = acc;
    }
    __syncthreads();  // all waves done reading before this buffer is re-filled
  }

  // epilogue: normalize, add residual, LayerNorm(row), store
  float s1[8], s2[8], invl[8];
  #pragma unroll
  for (int r = 0; r < 8; ++r) { invl[r] = 1.0f / l_r[r]; s1[r] = 0.f; s2[r] = 0.f; }
  #pragma unroll
  for (int t = 0; t < 16; ++t) {
    #pragma unroll
    for (int r = 0; r < 8; ++r) {
      float x = o[t][r] * invl[r]
              + R[(size_t)(row0 + r + 8 * half) * D_MODEL + t * 16 + l16];
      o[t][r] = x;
      s1[r] += x; s2[r] += x * x;
    }
  }
  #pragma unroll
  for (int r = 0; r < 8; ++r) {
    const float su = redsum16(s1[r]);
    const float sq = redsum16(s2[r]);
    const float mu = su * (1.0f / D_MODEL);
    const float var = sq * (1.0f / D_MODEL) - mu * mu;
    s1[r] = mu;
    s2[r] = rsqrtf(var + LN_EPS);
  }
  #pragma unroll
  for (int t = 0; t < 16; ++t) {
    const float g = gamma[t * 16 + l16], bb = beta[t * 16 + l16];
    #pragma unroll
    for (int r = 0; r < 8; ++r)
      out[(size_t)(row0 + r + 8 * half) * D_MODEL + t * 16 + l16] =
          (o[t][r] - s1[r]) * s2[r] * g + bb;
  }
}

// ---------------------------------------------------------------------------
// out = LayerNorm(relu(X@W1+b1)@W2 + b2 + X).  One 16-row strip per block.
// ---------------------------------------------------------------------------
__global__ __launch_bounds__(256) void mlp_ln_kernel(
    const float* __restrict__ X, const float* __restrict__ W1, const float* __restrict__ b1,
    const float* __restrict__ W2, const float* __restrict__ b2,
    const float* __restrict__ gamma, const float* __restrict__ beta,
    float* __restrict__ out)
{
  __shared__ float Xs[16 * D_MODEL];  // 16 KB
  __shared__ float Hs[16 * H_DIM];    //  8 KB
  __shared__ float Ys[16 * D_MODEL];  // 16 KB
  const int row0 = blockIdx.x * 16;
  {
    const float4* src = (const float4*)(X + (size_t)row0 * D_MODEL);
    float4* dst = (float4*)Xs;
    for (int i = threadIdx.x; i < 16 * D_MODEL / 4; i += blockDim.x) dst[i] = src[i];
  }
  __syncthreads();

  const int wave = threadIdx.x >> 5, lane = threadIdx.x & 31;
  const int half = lane >> 4, l16 = lane & 15;

  {  // H = relu(X@W1 + b1): wave -> one 16x16 tile of the 16x128 H
    const int n0 = wave * 16;
    v8f acc = {};
    #pragma unroll 8
    for (int kk = 0; kk < D_MODEL; kk += 4) {
      const int ka = kk + 2 * half;
      v2f a = *(const v2f*)&Xs[l16 * D_MODEL + ka];
      v2f b;
      b.x = W1[(size_t)ka * H_DIM + n0 + l16];
      b.y = W1[(size_t)(ka + 1) * H_DIM + n0 + l16];
      acc = wmma4(a, b, acc);
    }
    const float bv = b1[n0 + l16];
    #pragma unroll
    for (int r = 0; r < 8; ++r)
      Hs[(r + 8 * half) * H_DIM + n0 + l16] = fmaxf(acc[r] + bv, 0.0f);
  }
  __syncthreads();

  for (int t = 0; t < 2; ++t) {  // Y = H@W2 + b2 + X
    const int n0 = wave * 32 + t * 16;
    v8f acc = {};
    #pragma unroll 8
    for (int kk = 0; kk < H_DIM; kk += 4) {
      const int ka = kk + 2 * half;
      v2f a = *(const v2f*)&Hs[l16 * H_DIM + ka];
      v2f b;
      b.x = W2[(size_t)ka * D_MODEL + n0 + l16];
      b.y = W2[(size_t)(ka + 1) * D_MODEL + n0 + l16];
      acc = wmma4(a, b, acc);
    }
    const float bv = b2[n0 + l16];
    #pragma unroll
    for (int r = 0; r < 8; ++r) {
      const int m = r + 8 * half;
      Ys[m * D_MODEL + n0 + l16] = acc[r] + bv + Xs[m * D_MODEL + n0 + l16];
    }
  }
  __syncthreads();

  // LayerNorm: 16 threads per row, coalesced stores
  const int g = threadIdx.x >> 4, i = threadIdx.x & 15;
  float s1 = 0.f, s2 = 0.f;
  for (int c = i; c < D_MODEL; c += 16) { const float x = Ys[g * D_MODEL + c]; s1 += x; s2 += x * x; }
  s1 = redsum16(s1); s2 = redsum16(s2);
  const float mu = s1 * (1.0f / D_MODEL);
  const float rstd = rsqrtf(s2 * (1.0f / D_MODEL) - mu * mu + LN_EPS);
  for (int c = i; c < D_MODEL; c += 16)
    out[(size_t)(row0 + g) * D_MODEL + c] = (Ys[g * D_MODEL + c] - mu) * rstd * gamma[c] + beta[c];
}

// ---------------------------------------------------------------------------
extern "C" void kernel_launch(void* const* d_in, const int* in_sizes, int n_in,
                              void* d_out, int out_size, void* d_ws, size_t ws_size,
                              hipStream_t stream)
{
  const float* query1 = (const float*)d_in[0];
  const float* key    = (const float*)d_in[1];
  const float* value  = (const float*)d_in[2];
  const float* query2 = (const float*)d_in[3];
  const float* Wq = (const float*)d_in[4];
  const float* bq = (const float*)d_in[5];
  const float* Wk = (const float*)d_in[6];
  const float* bk = (const float*)d_in[7];
  const float* Wv = (const float*)d_in[8];
  const float* bv = (const float*)d_in[9];
  const float* W1 = (const float*)d_in[10];
  const float* b1 = (const float*)d_in[11];
  const float* W2 = (const float*)d_in[12];
  const float* b2 = (const float*)d_in[13];
  const float* gamma = (const float*)d_in[14];
  const float* beta  = (const float*)d_in[15];

  const int S = in_sizes[0] / D_MODEL;        // 8192
  const size_t SD = (size_t)S * D_MODEL;
  float* q = (float*)d_ws;                    // 4 x 8 MB workspace slots
  float* k = q + SD;
  float* v = k + SD;
  float* o = v + SD;

  dim3 gProj(S / 16), bProj(256);
  dim3 gAttn(S / 32), bAttn(64);

  // block 1
  proj_gemm_kernel<<<gProj, bProj, 0, stream>>>(query1, Wq, bq, q);
  proj_gemm_kernel<<<gProj, bProj, 0, stream>>>(key,    Wk, bk, k);
  proj_gemm_kernel<<<gProj, bProj, 0, stream>>>(value,  Wv, bv, v);
  attn_ln_kernel <<<gAttn, bAttn, 0, stream>>>(q, k, v, query1, gamma, beta, o, S);
  // block 2 (k2/v2 from out1 = o; then o is overwritten with out2)
  proj_gemm_kernel<<<gProj, bProj, 0, stream>>>(query2, Wq, bq, q);
  proj_gemm_kernel<<<gProj, bProj, 0, stream>>>(o, Wk, bk, k);
  proj_gemm_kernel<<<gProj, bProj, 0, stream>>>(o, Wv, bv, v);
  attn_ln_kernel <<<gAttn, bAttn, 0, stream>>>(q, k, v, query2, gamma, beta, o, S);
  // MLP + residual + LN -> output
  mlp_ln_kernel  <<<gProj, bProj, 0, stream>>>(o, W1, b1, W2, b2, gamma, beta, (float*)d_out);
}